// ARCEmbeddings_81458349736646
// MI455X (gfx1250) — compile-verified
//
#include <hip/hip_runtime.h>
#include <hip/hip_bf16.h>

// ---------------- constants from the reference ----------------
#define TOK_PAD 10
#define TOK_BOS 11
#define TOK_EOS 12
#define TOK_ROW 13
#define TOK_SEP 14
#define BB 16
#define TT 4096
#define DD 512
#define MAXRC 30
#define LN_EPS 1e-5f

typedef float v2f __attribute__((ext_vector_type(2)));
typedef float v8f __attribute__((ext_vector_type(8)));

// =====================================================================
// Kernel 1: parallel segmented scan producing packed (row<<8)|col ids.
// One wave (32 lanes) per sequence; each lane owns a 128-token chunk.
// Chunk transforms compose associatively -> Hillis-Steele wave scans.
// =====================================================================
__global__ __launch_bounds__(32)
void arc_scan_ids(const int* __restrict__ ids, unsigned* __restrict__ out_rc)
{
    __shared__ int toks[TT];
    const int b    = blockIdx.x;
    const int lane = threadIdx.x;          // 0..31 (wave32)
    const int* seq = ids + b * TT;
    for (int t = lane; t < TT; t += 32) toks[t] = seq[t];
    __syncthreads();

    const int CH   = TT / 32;              // 128 tokens per lane
    const int base = lane * CH;

    // ---- pass 1: per-chunk aggregates for in_grid and row ----
    int rReset = 0, rVal = 0;              // row: r_out = rReset ? rVal : r_in + rVal
    int gSet = 0, gVal = 0;                // grid: g_out = gSet ? gVal : g_in
    for (int i = 0; i < CH; ++i) {
        const int tk = toks[base + i];
        if (tk == TOK_BOS)                        { gSet = 1; gVal = 1; }
        else if (tk == TOK_EOS || tk == TOK_PAD)  { gSet = 1; gVal = 0; }
        if (tk == TOK_BOS || tk == TOK_SEP)       { rReset = 1; rVal = 0; }
        else if (tk == TOK_ROW)                   { rVal++; }
    }
    // inclusive wave scan (compose earlier-then-self)
    int iR = rReset, iV = rVal, iGS = gSet, iGV = gVal;
    for (int off = 1; off < 32; off <<= 1) {
        int eR  = __shfl_up(iR, off, 32);
        int eV  = __shfl_up(iV, off, 32);
        int eGS = __shfl_up(iGS, off, 32);
        int eGV = __shfl_up(iGV, off, 32);
        if (lane >= off) {
            iV  = iR  ? iV  : (eV + iV);
            iR  = iR  | eR;
            iGV = iGS ? iGV : eGV;
            iGS = iGS | eGS;
        }
    }
    // exclusive prefix applied to initial state (r=0, g=false)
    int xV  = __shfl_up(iV, 1, 32);
    int xGS = __shfl_up(iGS, 1, 32);
    int xGV = __shfl_up(iGV, 1, 32);
    const int r_in = (lane == 0) ? 0 : xV;            // reset? v : 0+v  == v
    const int g_in = (lane == 0) ? 0 : (xGS & xGV);   // gSet? gVal : false

    // ---- pass 2: col aggregate (g now fully determined per chunk) ----
    int cReset = 0, cVal = 0;
    {
        int g = g_in;
        for (int i = 0; i < CH; ++i) {
            const int tk    = toks[base + i];
            const int digit = (tk >= 0) & (tk <= 9);
            const int gc    = g & digit;
            if (tk == TOK_BOS || tk == TOK_SEP || tk == TOK_ROW) { cReset = 1; cVal = 0; }
            else if (gc)                                          { cVal++; }
            if (tk == TOK_BOS)                       g = 1;
            else if (tk == TOK_EOS || tk == TOK_PAD) g = 0;
        }
    }
    int iC = cReset, iCV = cVal;
    for (int off = 1; off < 32; off <<= 1) {
        int eC  = __shfl_up(iC, off, 32);
        int eCV = __shfl_up(iCV, off, 32);
        if (lane >= off) {
            iCV = iC ? iCV : (eCV + iCV);
            iC  = iC | eC;
        }
    }
    int xCV = __shfl_up(iCV, 1, 32);
    const int c_in = (lane == 0) ? 0 : xCV;

    // ---- pass 3: emit clipped, masked (row,col) per position ----
    {
        int r = r_in, c = c_in, g = g_in;
        unsigned* out = out_rc + b * TT;
        for (int i = 0; i < CH; ++i) {
            const int tk    = toks[base + i];
            const int digit = (tk >= 0) & (tk <= 9);
            const int gc    = g & digit;                    // uses state BEFORE update
            const int er    = gc ? ((r < MAXRC - 1) ? r : (MAXRC - 1)) : 0;
            const int ec    = gc ? ((c < MAXRC - 1) ? c : (MAXRC - 1)) : 0;
            out[base + i]   = ((unsigned)er << 8) | (unsigned)ec;
            if (tk == TOK_BOS)                       g = 1;
            else if (tk == TOK_EOS || tk == TOK_PAD) g = 0;
            if (tk == TOK_BOS || tk == TOK_SEP)      r = 0;
            else if (tk == TOK_ROW)                  r++;
            if (tk == TOK_BOS || tk == TOK_SEP || tk == TOK_ROW) c = 0;
            else if (gc)                                          c++;
        }
    }
}

// =====================================================================
// Kernel 2: Arow = row_table @ W[:, :256]^T  (30x512), Acol likewise,
// via V_WMMA_F32_16X16X4_F32 (exact f32). One wave per 16x16 D-tile.
// grid.x = 128: target(2) x Mtile(2, M padded 30->32) x Ntile(32).
// =====================================================================
__global__ __launch_bounds__(32)
void arc_precompute_wmma(const float* __restrict__ row_table,
                         const float* __restrict__ col_table,
                         const float* __restrict__ w_spatial,
                         float* __restrict__ Arow,
                         float* __restrict__ Acol)
{
    const int bid    = blockIdx.x;
    const int target = bid >> 6;        // 0 -> Arow, 1 -> Acol
    const int mt     = (bid >> 5) & 1;  // M tile
    const int nt     = bid & 31;        // N tile
    const float* A     = target ? col_table : row_table;
    const float* Wb    = w_spatial + (target ? 256 : 0);   // B[k][n] = W[n][off+k]
    float*       D     = target ? Acol : Arow;

    const int lane = threadIdx.x & 31;
    const int half = lane >> 4;         // K sub-pair select (ISA 16x4 A layout)
    const int lm   = lane & 15;
    const int m    = mt * 16 + lm;
    const int mA   = (m < MAXRC) ? m : 0;   // clamp padded rows (results discarded)
    const int n    = nt * 16 + lm;

    v8f acc = {};
    #pragma unroll 4
    for (int k0 = 0; k0 < 256; k0 += 4) {
        const int kb = k0 + 2 * half;
        v2f a = *(const v2f*)(A  + mA * 256 + kb);      // A[m][kb..kb+1]
        v2f bfrag = *(const v2f*)(Wb + n * 512 + kb);   // B[kb..kb+1][n] = W[n][off+kb..]
        acc = __builtin_amdgcn_wmma_f32_16x16x4_f32(
            /*neg_a=*/false, a, /*neg_b=*/false, bfrag,
            /*c_mod=*/(short)0, acc, /*reuse_a=*/false, /*reuse_b=*/false);
    }
    // C/D layout: lane holds N = lm; rows = mt*16 + i + 8*half, i = 0..7
    #pragma unroll
    for (int i = 0; i < 8; ++i) {
        const int mr = mt * 16 + i + 8 * half;
        if (mr < MAXRC) D[mr * DD + n] = acc[i];
    }
}

// =====================================================================
// Kernel 3: fused  x = tok_emb + pos_emb + mask*(Arow[r]+Acol[c]); LN.
// One wave per (b,t) row; float4 vectorized; shfl_xor reductions.
// =====================================================================
__global__ __launch_bounds__(256)
void arc_embed_ln(const int* __restrict__ ids,
                  const unsigned* __restrict__ rc,
                  const float* __restrict__ token_table,
                  const float* __restrict__ pos_table,
                  const float* __restrict__ Arow,
                  const float* __restrict__ Acol,
                  const float* __restrict__ gamma,
                  const float* __restrict__ beta,
                  float* __restrict__ out)
{
    const int gtid = blockIdx.x * blockDim.x + threadIdx.x;
    const int row  = gtid >> 5;                 // one wave32 per (b,t) row
    const int lane = threadIdx.x & 31;
    const int t    = row & (TT - 1);

    const int      tok = ids[row];
    const unsigned rcv = rc[row];
    const int r = (int)(rcv >> 8);
    const int c = (int)(rcv & 255u);
    const float mask = (tok >= 0 && tok <= 9) ? 1.0f : 0.0f;

    const float4* tokp = (const float4*)(token_table + (size_t)tok * DD);
    const float4* posp = (const float4*)(pos_table   + (size_t)t   * DD);
    const float4* arp  = (const float4*)(Arow        + (size_t)r   * DD);
    const float4* acp  = (const float4*)(Acol        + (size_t)c   * DD);

    float4 x[4];
    float s = 0.0f, s2 = 0.0f;
    #pragma unroll
    for (int i = 0; i < 4; ++i) {
        const int j = i * 32 + lane;            // covers 0..127 float4s = 512 floats
        const float4 tv = tokp[j];
        const float4 pv = posp[j];
        const float4 av = arp[j];
        const float4 cv = acp[j];
        float4 v;
        v.x = tv.x + pv.x + mask * (av.x + cv.x);
        v.y = tv.y + pv.y + mask * (av.y + cv.y);
        v.z = tv.z + pv.z + mask * (av.z + cv.z);
        v.w = tv.w + pv.w + mask * (av.w + cv.w);
        x[i] = v;
        s  += v.x + v.y + v.z + v.w;
        s2 += v.x * v.x + v.y * v.y + v.z * v.z + v.w * v.w;
    }
    #pragma unroll
    for (int off = 16; off > 0; off >>= 1) {
        s  += __shfl_xor(s,  off, 32);
        s2 += __shfl_xor(s2, off, 32);
    }
    const float mean = s * (1.0f / (float)DD);
    const float var  = s2 * (1.0f / (float)DD) - mean * mean;   // biased, like jnp.var
    const float inv  = rsqrtf(var + LN_EPS);

    float4*       op = (float4*)(out + (size_t)row * DD);
    const float4* gp = (const float4*)gamma;
    const float4* bp = (const float4*)beta;
    #pragma unroll
    for (int i = 0; i < 4; ++i) {
        const int j = i * 32 + lane;
        const float4 g4 = gp[j];
        const float4 b4 = bp[j];
        const float4 v  = x[i];
        float4 o;
        o.x = (v.x - mean) * inv * g4.x + b4.x;
        o.y = (v.y - mean) * inv * g4.y + b4.y;
        o.z = (v.z - mean) * inv * g4.z + b4.z;
        o.w = (v.w - mean) * inv * g4.w + b4.w;
        op[j] = o;
    }
}

// =====================================================================
extern "C" void kernel_launch(void* const* d_in, const int* in_sizes, int n_in,
                              void* d_out, int out_size, void* d_ws, size_t ws_size,
                              hipStream_t stream)
{
    (void)in_sizes; (void)n_in; (void)out_size; (void)ws_size;
    const int*   input_ids   = (const int*)  d_in[0];   // (B,T) int32
    const float* token_table = (const float*)d_in[1];   // (15,512)
    const float* pos_table   = (const float*)d_in[2];   // (4096,512)
    const float* row_table   = (const float*)d_in[3];   // (30,256)
    const float* col_table   = (const float*)d_in[4];   // (30,256)
    const float* w_spatial   = (const float*)d_in[5];   // (512,512)
    const float* ln_gamma    = (const float*)d_in[6];   // (512,)
    const float* ln_beta     = (const float*)d_in[7];   // (512,)
    float* out = (float*)d_out;                         // (B,T,512) f32

    // workspace layout
    char* ws = (char*)d_ws;
    unsigned* rc_ids = (unsigned*)ws;                                  // B*T u32   = 256 KiB
    float*    Arow   = (float*)(ws + (size_t)BB * TT * sizeof(unsigned)); // 30*512 f32
    float*    Acol   = Arow + MAXRC * DD;

    // 1) parallel 2D-id scan: one wave per sequence
    arc_scan_ids<<<BB, 32, 0, stream>>>(input_ids, rc_ids);

    // 2) WMMA precompute of the 30-row factored GEMMs
    arc_precompute_wmma<<<128, 32, 0, stream>>>(row_table, col_table, w_spatial,
                                                Arow, Acol);

    // 3) fused gather + add + mask + LayerNorm (memory-bound streaming pass)
    const int rows = BB * TT;                 // 65536 waves
    arc_embed_ln<<<rows * 32 / 256, 256, 0, stream>>>(
        input_ids, rc_ids, token_table, pos_table, Arow, Acol,
        ln_gamma, ln_beta, out);
}